// ModifiedChamferLoss_55155970015344
// MI455X (gfx1250) — compile-verified
//
#include <hip/hip_runtime.h>
#include <math.h>

typedef __attribute__((ext_vector_type(2))) float v2f;
typedef __attribute__((ext_vector_type(8))) float v8f;

#define HH 512
#define WW 512
#define MM 512
#define EPSF 1e-6f

// max_dist = sqrt(512^2 + 512^2)
__device__ __constant__ float kMaxD = 724.0773439350246f;
// EPS / max_dist
__device__ __constant__ float kEpsC = 1.3810957e-09f;

// ---------------- init workspace ----------------
__global__ void init_ws_kernel(unsigned int* __restrict__ colmin_bits,
                               float* __restrict__ sums) {
  int i = blockIdx.x * blockDim.x + threadIdx.x;
  if (i < MM) colmin_bits[i] = 0x7f800000u;  // +inf
  if (i < 2)  sums[i] = 0.0f;                // sums[0]=sum_p, sums[1]=sum_p*d
}

// ---------------- main tile kernel ----------------
// Each wave computes 16(pixel)x16(gt) distance tiles with V_WMMA_F32_16X16X4_F32:
//   A[i,:] = [y_i, x_i, 1, 0]   (lanes 0-15: K=0,1 ; lanes 16-31: K=2,3)
//   B[:,j] = [-2*gy_j, -2*gx_j, |g_j|^2, 0]
//   d2[i,j] = D[i,j] + (y_i^2 + x_i^2)
// B operands are staged in LDS in lane-ready layout so the inner loop does a
// single unconditional ds_load_b64 (immediate offset) per tile. The mt loop is
// kept rolled (unroll 4): full unrolling makes the scheduler cluster all 32
// WMMA accumulators and blow past VGPR 255 (s_set_vgpr_msb thrash).
__global__ __launch_bounds__(256) void chamfer_tiles_kernel(
    const float* __restrict__ prob, const float* __restrict__ gt,
    unsigned int* __restrict__ colmin_bits, float* __restrict__ sums) {
  __shared__ v2f Bs[32 * 32];        // [mt][lane] lane-ready B operands (8KB)
  __shared__ unsigned int cm[MM];    // per-block column-min (float bits)
  __shared__ float sred[2][8];

  const int tid  = threadIdx.x;
  const int lane = tid & 31;
  const int wave = tid >> 5;

  // stage lane-ready B operands + init block colmin
  for (int e = tid; e < 32 * 32; e += 256) {
    const int mt    = e >> 5;
    const int l15e  = e & 15;
    const int halfe = (e >> 4) & 1;
    const int m     = (mt << 4) + l15e;
    const float gy = gt[2 * m + 0];
    const float gx = gt[2 * m + 1];
    v2f b;
    if (halfe) { b.x = gy * gy + gx * gx; b.y = 0.0f; }        // K=2,3
    else       { b.x = -2.0f * gy;        b.y = -2.0f * gx; }  // K=0,1
    Bs[e] = b;
  }
  for (int m = tid; m < MM; m += 256) cm[m] = 0x7f800000u;
  __syncthreads();

  const int half = lane >> 4;        // 0: rows 0-7 / K=0,1 ; 1: rows 8-15 / K=2,3
  const int l15  = lane & 15;
  const int off  = half << 3;        // 0 or 8

  float sum_p_loc  = 0.0f;
  float sum_pd_loc = 0.0f;

  const int tile0 = blockIdx.x * 64 + wave * 8;  // 256 blocks * 64 tiles = 16384

#pragma unroll 1
  for (int t = 0; t < 8; ++t) {
    const int n0 = (tile0 + t) << 4;         // 16 consecutive pixels, same row
    const int y  = n0 >> 9;                  // n / 512
    const int xb = n0 & (WW - 1);
    const float yf = (float)y;

    // A operand (16x4 f32 -> 2 VGPRs/lane)
    v2f A;
    A.x = half ? 1.0f : yf;
    A.y = half ? 0.0f : (float)(xb + lane);

    // per-lane prob + inverse-p^4 weight for one pixel of this tile
    const float pj = prob[n0 + l15];
    if (!half) sum_p_loc += pj;              // count each pixel once
    const float p2 = pj * pj;
    const float wj = __builtin_amdgcn_rcpf(p2 * p2 + kEpsC);  // v_rcp_f32

    // broadcast this half's 8 weights / eps-weights / pixel-norms
    float wr[8], ewr[8], cr[8];
#pragma unroll
    for (int r = 0; r < 8; ++r) {
      wr[r]  = __shfl(wj, r + off, 32);
      ewr[r] = EPSF * wr[r];
      const float xr = (float)(xb + off + r);
      cr[r] = yf * yf + xr * xr;
    }

    float rowmin2[8];
#pragma unroll
    for (int r = 0; r < 8; ++r) rowmin2[r] = __builtin_inff();

#pragma unroll 4
    for (int mt = 0; mt < 32; ++mt) {
      // single b64 LDS load, constant offset mt*256 per unrolled iteration
      const v2f B = Bs[(mt << 5) + lane];
      v8f C = {};
      // D = A x B  (cross + |g|^2 terms of squared distance)
      v8f D = __builtin_amdgcn_wmma_f32_16x16x4_f32(
          false, A, false, B, (short)0, C, false, false);

      float colv = __builtin_inff();
#pragma unroll
      for (int r = 0; r < 8; ++r) {
        float d2 = D[r] + cr[r];
        d2 = fmaxf(d2, 0.0f);
        rowmin2[r] = fminf(rowmin2[r], d2);
        // (sqrt(d2)+EPS)*wr  ==  fma(sqrt(d2), wr, EPS*wr); raw v_sqrt_f32
        const float v = __builtin_fmaf(__builtin_amdgcn_sqrtf(d2), wr[r], ewr[r]);
        colv = fminf(colv, v);
      }
      // lanes L and L+16 hold the same m with disjoint pixel subsets;
      // both contribute partial column mins (values >= 0 -> uint-bit order ok)
      atomicMin(&cm[(mt << 4) + l15], __float_as_uint(colv));
    }

    // row-min: reduce across the 16 lanes of each half
#pragma unroll
    for (int mask = 1; mask <= 8; mask <<= 1) {
#pragma unroll
      for (int r = 0; r < 8; ++r)
        rowmin2[r] = fminf(rowmin2[r], __shfl_xor(rowmin2[r], mask, 32));
    }
    if (l15 == 0) {  // lanes 0 (pixels n0..n0+7) and 16 (pixels n0+8..n0+15)
#pragma unroll
      for (int r = 0; r < 8; ++r) {
        const int n = n0 + off + r;
        sum_pd_loc += prob[n] * __builtin_amdgcn_sqrtf(rowmin2[r]);
      }
    }
  }

  // wave-level sum reduction
#pragma unroll
  for (int mask = 16; mask >= 1; mask >>= 1) {
    sum_p_loc  += __shfl_xor(sum_p_loc,  mask, 32);
    sum_pd_loc += __shfl_xor(sum_pd_loc, mask, 32);
  }
  if (lane == 0) { sred[0][wave] = sum_p_loc; sred[1][wave] = sum_pd_loc; }
  __syncthreads();

  // flush block column-mins to global
  for (int m = tid; m < MM; m += 256)
    atomicMin(&colmin_bits[m], cm[m]);

  if (tid == 0) {
    float sp = 0.0f, spd = 0.0f;
#pragma unroll
    for (int i = 0; i < 8; ++i) { sp += sred[0][i]; spd += sred[1][i]; }
    atomicAdd(&sums[0], sp);
    atomicAdd(&sums[1], spd);
  }
}

// ---------------- finalize ----------------
__global__ __launch_bounds__(512) void finalize_kernel(
    const unsigned int* __restrict__ colmin_bits,
    const float* __restrict__ sums, float* __restrict__ out) {
  __shared__ float red[16];
  const int tid = threadIdx.x;
  float v = __uint_as_float(colmin_bits[tid]);
  v = fminf(fmaxf(v, 0.0f), kMaxD);  // clip to [0, max_dist]
#pragma unroll
  for (int mask = 16; mask >= 1; mask >>= 1) v += __shfl_xor(v, mask, 32);
  if ((tid & 31) == 0) red[tid >> 5] = v;
  __syncthreads();
  if (tid == 0) {
    float s = 0.0f;
#pragma unroll
    for (int i = 0; i < 16; ++i) s += red[i];
    const float term2 = s / (float)MM;
    const float term1 = sums[1] / (sums[0] + EPSF);
    out[0] = term1 + term2;
  }
}

extern "C" void kernel_launch(void* const* d_in, const int* in_sizes, int n_in,
                              void* d_out, int out_size, void* d_ws, size_t ws_size,
                              hipStream_t stream) {
  const float* prob = (const float*)d_in[0];   // [512*512]
  const float* gt   = (const float*)d_in[1];   // [512,2]
  // d_in[2] (all_img_locations) is unused: locations derived analytically.

  unsigned int* colmin_bits = (unsigned int*)d_ws;
  float* sums = (float*)((char*)d_ws + MM * sizeof(unsigned int));

  init_ws_kernel<<<2, 256, 0, stream>>>(colmin_bits, sums);
  chamfer_tiles_kernel<<<256, 256, 0, stream>>>(prob, gt, colmin_bits, sums);
  finalize_kernel<<<1, 512, 0, stream>>>(colmin_bits, sums, (float*)d_out);
}